// GCN_47605417509108
// MI455X (gfx1250) — compile-verified
//
#include <hip/hip_runtime.h>

// GCN (2-layer) for MI455X / gfx1250.
// Pipeline:
//   K0 deg_init      : deg[i] = 1 (self-loop)
//   K1 deg_edges     : deg[dst] += 1 (atomic)
//   K2 dinv_self     : dinv = rsqrt(deg);  s[i] = dinv^2 * x[i]   (layer-1 self-loop)
//   K3 agg1          : s[dst] += dinv[src]*dinv[dst]*x[src]       (scalar messages, C_IN==1)
//   K4 transform     : g[i] = relu(s[i]*W1 + b1) @ W2  via V_WMMA_F32_16X16X4_F32,
//                      out[i] = b2 + dinv[i]^2 * g[i]             (layer-2 self-loop + bias)
//   K5 agg2          : out[dst] += dinv[src]*dinv[dst]*g[src]     (8B messages)
//
// Workspace (floats): [0,N) dinv/deg, [N,2N) s, [2N,4N) g  -> 1.6 MB, L2-resident.

#define GCN_H     16
#define GCN_COUT  2

typedef __attribute__((ext_vector_type(2))) float v2f;
typedef __attribute__((ext_vector_type(8))) float v8f;

__global__ void gcn_deg_init(float* __restrict__ deg, int N) {
  int i = blockIdx.x * blockDim.x + threadIdx.x;
  if (i < N) deg[i] = 1.0f;  // self-loop contributes 1 to every node's in-degree
}

__global__ void gcn_deg_edges(const int* __restrict__ dst, float* __restrict__ deg, int E) {
  int e = blockIdx.x * blockDim.x + threadIdx.x;
  if (e < E) atomicAdd(&deg[dst[e]], 1.0f);
}

__global__ void gcn_dinv_self(const float* __restrict__ x, float* __restrict__ dinv,
                              float* __restrict__ s, int N) {
  int i = blockIdx.x * blockDim.x + threadIdx.x;
  if (i < N) {
    float r = rsqrtf(dinv[i]);   // buffer currently holds deg >= 1
    dinv[i] = r;
    s[i] = r * r * x[i];         // self-loop message, layer 1
  }
}

__global__ void gcn_agg1(const int* __restrict__ src, const int* __restrict__ dst,
                         const float* __restrict__ x, const float* __restrict__ dinv,
                         float* __restrict__ s, int E) {
  int e = blockIdx.x * blockDim.x + threadIdx.x;
  if (e < E) {
    int u = src[e], v = dst[e];
    atomicAdd(&s[v], dinv[u] * dinv[v] * x[u]);
  }
}

// One wave per 16-node tile. D = A(16x16 h-tile, fed as 4 K=4 slices) x B(W2 padded 16x16).
// B-operand loads are branch-free: column index clamped in-bounds, padded columns zeroed
// with an arithmetic select so EXEC stays all-ones with no save/restore around the WMMAs.
__global__ void gcn_transform_wmma(const float* __restrict__ s, const float* __restrict__ dinv,
                                   const float* __restrict__ W1, const float* __restrict__ b1,
                                   const float* __restrict__ W2, const float* __restrict__ b2,
                                   float* __restrict__ g, float* __restrict__ out, int nTiles) {
  int wave = (int)((blockIdx.x * blockDim.x + threadIdx.x) >> 5);
  int lane = threadIdx.x & 31;
  if (wave >= nTiles) return;           // wave-uniform: EXEC all-ones for WMMA
  int m   = lane & 15;                  // A: row M; B/D: column N
  int hi  = lane >> 4;                  // half-wave select (K or M offset)
  int row0 = wave << 4;

  int   mc      = m & 1;                            // clamped (always in-bounds) column
  float colMask = (m < GCN_COUT) ? 1.0f : 0.0f;     // v_cndmask, no branch

  float sv  = s[row0 + m];              // node scalar for row m of this tile
  float b2v = b2[mc];                   // only consumed by lanes with m < COUT

  v8f c = {};
#pragma unroll
  for (int k0 = 0; k0 < GCN_H; k0 += 4) {
    int ka = k0 + 2 * hi;               // lanes 0-15 hold K={k0,k0+1}, lanes 16-31 K={k0+2,k0+3}
    v2f a, b;
    // A element = relu(s*W1 + b1) computed on the fly (fused transform)
    a.x = fmaxf(fmaf(sv, W1[ka],     b1[ka]),     0.0f);
    a.y = fmaxf(fmaf(sv, W1[ka + 1], b1[ka + 1]), 0.0f);
    // B = W2 zero-padded to 16 columns; unconditional in-bounds load + mask
    b.x = W2[ka * GCN_COUT + mc]       * colMask;
    b.y = W2[(ka + 1) * GCN_COUT + mc] * colMask;
    c = __builtin_amdgcn_wmma_f32_16x16x4_f32(
        /*neg_a=*/false, a, /*neg_b=*/false, b,
        /*c_mod=*/(short)0, c, /*reuse_a=*/false, /*reuse_b=*/false);
  }

  // D layout: VGPR j, lanes 0-15 -> M=j, N=lane; lanes 16-31 -> M=j+8, N=lane-16.
  if (m < GCN_COUT) {
#pragma unroll
    for (int j = 0; j < 8; ++j) {
      int row  = row0 + j + 8 * hi;
      float gv = c[j];
      g[row * GCN_COUT + m] = gv;
      float r = dinv[row];
      out[row * GCN_COUT + m] = b2v + r * r * gv;  // bias + layer-2 self-loop
    }
  }
}

// Scalar fallback for rows not covered by full 16-row tiles (N % 16 != 0).
__global__ void gcn_transform_tail(const float* __restrict__ s, const float* __restrict__ dinv,
                                   const float* __restrict__ W1, const float* __restrict__ b1,
                                   const float* __restrict__ W2, const float* __restrict__ b2,
                                   float* __restrict__ g, float* __restrict__ out,
                                   int start, int N) {
  int i = start + blockIdx.x * blockDim.x + threadIdx.x;
  if (i >= N) return;
  float sv = s[i];
  float a0 = 0.0f, a1 = 0.0f;
#pragma unroll
  for (int k = 0; k < GCN_H; ++k) {
    float h = fmaxf(fmaf(sv, W1[k], b1[k]), 0.0f);
    a0 = fmaf(h, W2[k * GCN_COUT + 0], a0);
    a1 = fmaf(h, W2[k * GCN_COUT + 1], a1);
  }
  float r = dinv[i];
  g[i * GCN_COUT + 0] = a0;
  g[i * GCN_COUT + 1] = a1;
  out[i * GCN_COUT + 0] = b2[0] + r * r * a0;
  out[i * GCN_COUT + 1] = b2[1] + r * r * a1;
}

__global__ void gcn_agg2(const int* __restrict__ src, const int* __restrict__ dst,
                         const float* __restrict__ g, const float* __restrict__ dinv,
                         float* __restrict__ out, int E) {
  int e = blockIdx.x * blockDim.x + threadIdx.x;
  if (e < E) {
    int u = src[e], v = dst[e];
    float w  = dinv[u] * dinv[v];
    float g0 = g[u * GCN_COUT + 0];
    float g1 = g[u * GCN_COUT + 1];
    atomicAdd(&out[v * GCN_COUT + 0], w * g0);
    atomicAdd(&out[v * GCN_COUT + 1], w * g1);
  }
}

extern "C" void kernel_launch(void* const* d_in, const int* in_sizes, int n_in,
                              void* d_out, int out_size, void* d_ws, size_t ws_size,
                              hipStream_t stream) {
  const float* x  = (const float*)d_in[0];   // [N,1]
  const int*   ei = (const int*)d_in[1];     // [2,E]
  const float* W1 = (const float*)d_in[2];   // [1,16]
  const float* b1 = (const float*)d_in[3];   // [16]
  const float* W2 = (const float*)d_in[4];   // [16,2]
  const float* b2 = (const float*)d_in[5];   // [2]
  float* out = (float*)d_out;                // [N,2]

  const int N = in_sizes[0];                 // C_IN == 1
  const int E = in_sizes[1] / 2;
  const int* src = ei;
  const int* dst = ei + E;

  float* ws   = (float*)d_ws;
  float* dinv = ws;                          // also holds deg during passes 0-1
  float* s    = ws + (size_t)N;
  float* g    = ws + 2 * (size_t)N;

  const int TB = 256;
  const int gN = (N + TB - 1) / TB;
  const int gE = (E + TB - 1) / TB;

  gcn_deg_init  <<<gN, TB, 0, stream>>>(dinv, N);
  gcn_deg_edges <<<gE, TB, 0, stream>>>(dst, dinv, E);
  gcn_dinv_self <<<gN, TB, 0, stream>>>(x, dinv, s, N);
  gcn_agg1      <<<gE, TB, 0, stream>>>(src, dst, x, dinv, s, E);

  const int nTiles = N / 16;                 // N = 100000 -> 6250 full tiles
  if (nTiles > 0) {
    const int waves  = nTiles;               // one wave per tile
    const int blocks = (waves * 32 + TB - 1) / TB;
    gcn_transform_wmma<<<blocks, TB, 0, stream>>>(s, dinv, W1, b1, W2, b2, g, out, nTiles);
  }
  const int tail = N - nTiles * 16;
  if (tail > 0) {
    gcn_transform_tail<<<(tail + TB - 1) / TB, TB, 0, stream>>>(s, dinv, W1, b1, W2, b2, g, out,
                                                                nTiles * 16, N);
  }

  gcn_agg2<<<gE, TB, 0, stream>>>(src, dst, g, dinv, out, E);
}